// Simulator_21835613733065
// MI455X (gfx1250) — compile-verified
//
#include <hip/hip_runtime.h>

// ---------------------------------------------------------------------------
// 4R3C thermal-network Euler scan, parallelized as a blocked affine scan.
//   x_{t+1} = M x_t + v_t,   M = I + dt*A (3x3),  v_t = (dt*B) u_t (3-vec)
// Phase KV : v_t = N u_t for all t via V_WMMA_F32_16X16X4_F32 (full fp32 WMMA),
//            16 tiles per wave, prefetching the u stream.
// Phase K2 : per-block (L=256) affine carries
// Phase K2b: single-wave hierarchical scan of 4096 block carries
// Phase K3 : per-block replay producing xsol/ysol
// ---------------------------------------------------------------------------

typedef __attribute__((ext_vector_type(2))) float v2f;
typedef __attribute__((ext_vector_type(8))) float v8f;

#define T_STEPS (1 << 20)
#define BLK_L   256
#define NBLK    (T_STEPS / BLK_L)   // 4096 blocks
#define SEG     128                 // blocks per scan lane
#define NLANE   32                  // NBLK / SEG

#define TILES_TOTAL    (T_STEPS / 16)                 // 65536 WMMA tiles
#define TILES_PER_WAVE 16
#define KV_BLOCK       256
#define KV_GRID        ((TILES_TOTAL / TILES_PER_WAVE) / (KV_BLOCK / 32))  // 512

__device__ __forceinline__ void build_MN(const float* __restrict__ rc,
                                         float M[3][3], float N[3][5]) {
  const float DT = 3600.0f;
  float Rg = rc[0] * 10.0f + 1.0f;
  float Ri = rc[1] * 10.0f + 1.0f;
  float Re = rc[2] * 10.0f + 1.0f;
  float Rw = rc[3] * 10.0f + 1.0f;
  float Cai = rc[4] * 1.0e6f + 1.0e5f;
  float Cwe = rc[5] * 1.0e6f + 1.0e5f;
  float Cwi = rc[6] * 1.0e6f + 1.0e5f;
  float A00 = -(1.0f / Rg + 1.0f / Ri) / Cai;
  float A02 = 1.0f / (Ri * Cai);
  float A11 = -(1.0f / Re + 1.0f / Rw) / Cwe;
  float A12 = 1.0f / (Rw * Cwe);
  float A20 = 1.0f / (Ri * Cwi);
  float A21 = 1.0f / (Rw * Cwi);
  float A22 = -(1.0f / Rw + 1.0f / Ri) / Cwi;
  M[0][0] = 1.0f + DT * A00; M[0][1] = 0.0f;            M[0][2] = DT * A02;
  M[1][0] = 0.0f;            M[1][1] = 1.0f + DT * A11; M[1][2] = DT * A12;
  M[2][0] = DT * A20;        M[2][1] = DT * A21;        M[2][2] = 1.0f + DT * A22;
  N[0][0] = DT / (Rg * Cai); N[0][1] = DT / Cai; N[0][2] = DT / Cai;
  N[0][3] = 0.0f;            N[0][4] = 0.0f;
  N[1][0] = DT / (Re * Cwe); N[1][1] = 0.0f;     N[1][2] = 0.0f;
  N[1][3] = DT / Cwe;        N[1][4] = 0.0f;
  N[2][0] = 0.0f;            N[2][1] = 0.0f;     N[2][2] = 0.0f;
  N[2][3] = 0.0f;            N[2][4] = DT / Cwi;
}

__device__ __forceinline__ void mat3_mul(const float A[3][3], const float B[3][3],
                                         float C[3][3]) {
#pragma unroll
  for (int i = 0; i < 3; ++i)
#pragma unroll
    for (int j = 0; j < 3; ++j)
      C[i][j] = fmaf(A[i][0], B[0][j], fmaf(A[i][1], B[1][j], A[i][2] * B[2][j]));
}

__device__ __forceinline__ void mat3_sq_times(float M[3][3], int times) {
  for (int s = 0; s < times; ++s) {
    float Tm[3][3];
    mat3_mul(M, M, Tm);
#pragma unroll
    for (int i = 0; i < 3; ++i)
#pragma unroll
      for (int j = 0; j < 3; ++j) M[i][j] = Tm[i][j];
  }
}

// x <- M x + v
__device__ __forceinline__ void aff_step(const float M[3][3],
                                         float& x0, float& x1, float& x2,
                                         float v0, float v1, float v2) {
  float n0 = fmaf(M[0][0], x0, fmaf(M[0][1], x1, fmaf(M[0][2], x2, v0)));
  float n1 = fmaf(M[1][0], x0, fmaf(M[1][1], x1, fmaf(M[1][2], x2, v1)));
  float n2 = fmaf(M[2][0], x0, fmaf(M[2][1], x1, fmaf(M[2][2], x2, v2)));
  x0 = n0; x1 = n1; x2 = n2;
}

// --------------------------------------------------------------------------
// KV: v_t = N u_t, 16 timesteps per fp32 WMMA, 16 WMMA tiles per wave.
//   D(16x16) = A(16x4) x B(4x16):  A rows 0..2 = N[:,0:4],  B[k][n] = u[t0+n][k]
//   5th input channel handled with a VALU FMA.  v staged in the xsol region.
// Operand layouts per CDNA5 ISA 7.12.2 (32-bit A 16x4 / B 4x16, 2 VGPRs each).
// --------------------------------------------------------------------------
__global__ void kv_inject(const float* __restrict__ u, const float* __restrict__ rc,
                          float* __restrict__ vbuf) {
  float Mm[3][3], Nn[3][5];
  build_MN(rc, Mm, Nn);

  int lane = threadIdx.x & 31;
  int wave = (blockIdx.x * blockDim.x + threadIdx.x) >> 5;
  int half = lane >> 4;     // VGPR half selects K (B) / K-pair (A) per ISA layout
  int idx  = lane & 15;

  // A operand: rows 0..2 carry N[:,0:4], rest zero.  Invariant across tiles.
  v2f a;
  a.x = (idx < 3) ? Nn[idx][half * 2 + 0] : 0.0f;
  a.y = (idx < 3) ? Nn[idx][half * 2 + 1] : 0.0f;
  float n04 = Nn[0][4], n14 = Nn[1][4], n24 = Nn[2][4];

  int tile0 = wave * TILES_PER_WAVE;
#pragma unroll 4
  for (int it = 0; it < TILES_PER_WAVE; ++it) {
    int t0 = (tile0 + it) * 16;
    const float* up = u + (size_t)(t0 + idx) * 5;
    if (it + 1 < TILES_PER_WAVE)
      __builtin_prefetch(up + 16 * 5, 0, 0);   // global_prefetch_b8: next tile

    // B operand: column n = timestep t0+idx, K = u channel.
    v2f b;
    b.x = up[half * 2 + 0];
    b.y = up[half * 2 + 1];

    v8f c = {};
    v8f d = __builtin_amdgcn_wmma_f32_16x16x4_f32(
        /*neg_a=*/false, a, /*neg_b=*/false, b,
        /*c_mod=*/(short)0, c, /*reuse_a=*/false, /*reuse_b=*/false);

    // D rows 0..2 live in VGPRs 0..2 of lanes 0-15 (col = lane).
    if (half == 0) {
      float u4 = up[4];
      float* vp = vbuf + (size_t)(t0 + idx) * 3;
      vp[0] = fmaf(n04, u4, d[0]);
      vp[1] = fmaf(n14, u4, d[1]);
      vp[2] = fmaf(n24, u4, d[2]);
    }
  }
}

// --------------------------------------------------------------------------
// K2: per-block carry  c_b = fold_{j=0..L-1} ( c <- M c + v_{bL+j} ), c0 = 0
// --------------------------------------------------------------------------
__global__ void k2_carry(const float* __restrict__ vbuf, const float* __restrict__ rc,
                         float* __restrict__ cbuf) {
  float Mm[3][3], Nn[3][5];
  build_MN(rc, Mm, Nn);
  int b = blockIdx.x * blockDim.x + threadIdx.x;      // 0..NBLK-1
  const float* vp = vbuf + (size_t)b * BLK_L * 3;
  float c0 = 0.0f, c1 = 0.0f, c2 = 0.0f;
  for (int j = 0; j < BLK_L; ++j) {
    float v0 = vp[0], v1 = vp[1], v2 = vp[2];
    vp += 3;
    aff_step(Mm, c0, c1, c2, v0, v1, v2);
  }
  cbuf[(size_t)b * 3 + 0] = c0;
  cbuf[(size_t)b * 3 + 1] = c1;
  cbuf[(size_t)b * 3 + 2] = c2;
}

// --------------------------------------------------------------------------
// K2b: single-wave hierarchical scan of the 4096 block carries.
//   X_{b+1} = ML X_b + c_b with ML = M^256.
//   lane i folds its 128-block segment, lane 0 scans 32 segments with
//   Q = ML^128, lanes re-expand to all block-start states xb[b].
// --------------------------------------------------------------------------
__global__ void k2b_scan(const float* __restrict__ x_init, const float* __restrict__ rc,
                         const float* __restrict__ cbuf, float* __restrict__ xbbuf) {
  __shared__ float seg[NLANE][3];
  __shared__ float pre[NLANE][3];

  float Mm[3][3], Nn[3][5];
  build_MN(rc, Mm, Nn);
  float ML[3][3];
#pragma unroll
  for (int i = 0; i < 3; ++i)
#pragma unroll
    for (int j = 0; j < 3; ++j) ML[i][j] = Mm[i][j];
  mat3_sq_times(ML, 8);   // ML = M^256

  int lane = threadIdx.x;  // 0..31

  // segment fold
  float s0 = 0.0f, s1 = 0.0f, s2 = 0.0f;
  const float* cp = cbuf + (size_t)lane * SEG * 3;
  for (int j = 0; j < SEG; ++j) {
    aff_step(ML, s0, s1, s2, cp[0], cp[1], cp[2]);
    cp += 3;
  }
  seg[lane][0] = s0; seg[lane][1] = s1; seg[lane][2] = s2;
  __syncthreads();

  if (lane == 0) {
    float Q[3][3];
#pragma unroll
    for (int i = 0; i < 3; ++i)
#pragma unroll
      for (int j = 0; j < 3; ++j) Q[i][j] = ML[i][j];
    mat3_sq_times(Q, 7);   // Q = ML^128 = M^(256*128)
    float p0 = x_init[0], p1 = x_init[1], p2 = x_init[2];
    for (int i = 0; i < NLANE; ++i) {
      pre[i][0] = p0; pre[i][1] = p1; pre[i][2] = p2;
      aff_step(Q, p0, p1, p2, seg[i][0], seg[i][1], seg[i][2]);
    }
  }
  __syncthreads();

  // re-expand: block-start states within this lane's segment
  float x0 = pre[lane][0], x1 = pre[lane][1], x2 = pre[lane][2];
  const float* cp2 = cbuf + (size_t)lane * SEG * 3;
  float* xp = xbbuf + (size_t)lane * SEG * 3;
  for (int j = 0; j < SEG; ++j) {
    xp[0] = x0; xp[1] = x1; xp[2] = x2;
    xp += 3;
    aff_step(ML, x0, x1, x2, cp2[0], cp2[1], cp2[2]);
    cp2 += 3;
  }
}

// --------------------------------------------------------------------------
// K3: replay each block from its start state.
//   ysol[t] = x_t[0] (pre-update), xsol[t] = x_{t+1}.
//   v_t currently lives in the xsol slot; read it before overwriting.
// --------------------------------------------------------------------------
__global__ void k3_replay(const float* __restrict__ rc, const float* __restrict__ xbbuf,
                          float* __restrict__ out) {
  float Mm[3][3], Nn[3][5];
  build_MN(rc, Mm, Nn);
  int b = blockIdx.x * blockDim.x + threadIdx.x;      // 0..NBLK-1
  float x0 = xbbuf[(size_t)b * 3 + 0];
  float x1 = xbbuf[(size_t)b * 3 + 1];
  float x2 = xbbuf[(size_t)b * 3 + 2];
  float* xs = out + (size_t)b * BLK_L * 3;                     // xsol rows (holds v now)
  float* ys = out + (size_t)3 * T_STEPS + (size_t)b * BLK_L;   // ysol
  for (int j = 0; j < BLK_L; ++j) {
    float v0 = xs[0], v1 = xs[1], v2 = xs[2];   // read v_t before overwrite
    ys[j] = x0;                                 // y_t from pre-update state
    aff_step(Mm, x0, x1, x2, v0, v1, v2);
    xs[0] = x0; xs[1] = x1; xs[2] = x2;         // xsol[t] = x_{t+1}
    xs += 3;
  }
}

extern "C" void kernel_launch(void* const* d_in, const int* in_sizes, int n_in,
                              void* d_out, int out_size, void* d_ws, size_t ws_size,
                              hipStream_t stream) {
  const float* x_init = (const float*)d_in[0];   // [3]
  const float* u      = (const float*)d_in[1];   // [T,5]
  const float* rc     = (const float*)d_in[2];   // [7]
  float* out = (float*)d_out;                    // xsol [T,3] ++ ysol [T]

  float* cbuf  = (float*)d_ws;                   // NBLK*3 floats  (48 KB)
  float* xbbuf = cbuf + (size_t)NBLK * 3;        // NBLK*3 floats  (48 KB)
  float* vbuf  = out;                            // stage v_t in the xsol region

  // KV: 65536 tiles of 16 steps, 16 tiles per wave, 8 waves/block -> 512 blocks
  kv_inject<<<dim3(KV_GRID), dim3(KV_BLOCK), 0, stream>>>(u, rc, vbuf);
  // K2: one thread per 256-step block
  k2_carry<<<dim3(NBLK / 256), dim3(256), 0, stream>>>(vbuf, rc, cbuf);
  // K2b: single wave hierarchical scan
  k2b_scan<<<dim3(1), dim3(NLANE), 0, stream>>>(x_init, rc, cbuf, xbbuf);
  // K3: replay
  k3_replay<<<dim3(NBLK / 256), dim3(256), 0, stream>>>(rc, xbbuf, out);
}